// Topological_LED_Encoder_32908039422224
// MI455X (gfx1250) — compile-verified
//
#include <hip/hip_runtime.h>

// Topological LED encoder for MI455X (gfx1250, wave32, WMMA).
//
// The O(N^2)=67M pair phase dominates (~6 GFLOP total, all data L2-resident,
// HBM traffic negligible at 23.3 TB/s -> compute bound). It is mapped onto
// v_wmma_f32_16x16x4_f32:
//   (a) dist^2 tiles via |pj|^2 - 2 pj.pi (+|pi|^2 folded into the threshold):
//       256 pairs per WMMA.
//   (b) masked aggregation G = Y^T @ Adj (adjacency is symmetric, so the mask
//       tile -- already in WMMA C-layout from (a) -- becomes the B operand
//       after one SWAPX16 half-swap + half-select per register).
// Pass 1 emits a 1-bit/pair adjacency bitmap (8 MB, L2-resident) that pass 2
// (8-feature aggregation of z = h@W2+b2) consumes instead of recomputing
// distances. Y1 and Z are stored transposed (f-major) so each lane's two
// K-values for the A operand are contiguous -> global_load_b64.

typedef float v2f __attribute__((ext_vector_type(2)));
typedef float v8f __attribute__((ext_vector_type(8)));

#define N_NODES 8192
#define NTILES  512            // N_NODES / 16

// d_ws layout (bytes); total ~9.4 MB
#define WS_P4   (0)                                            // 8192 * float4
#define WS_Y1T  (131072)                                       // 32 x 8192 f32 (f-major)
#define WS_ZT   (131072 + 1048576)                             // 8 x 8192 f32 (f-major)
#define WS_CNT  (131072 + 1048576 + 262144)                    // 8192 f32
#define WS_BM   (131072 + 1048576 + 262144 + 32768)            // 2M dwords

static __device__ __forceinline__ float relu_f(float x) { return x > 0.f ? x : 0.f; }

// Swap lane halves (lane <-> lane^16). ds_swizzle BITMODE: and=0x1f, or=0,
// xor=0x10 -> offset 0x401F (SWAPX16).
static __device__ __forceinline__ float swap16(float x) {
#if __has_builtin(__builtin_amdgcn_ds_swizzle)
  return __int_as_float(__builtin_amdgcn_ds_swizzle(__float_as_int(x), 0x401F));
#else
  return __shfl_xor(x, 16, 32);
#endif
}

// ---------------------------------------------------------------------------
// Kernel 1: per-node features. P4[i] = (x,y,z,|p|^2); Y1T[f][i] = ([sin|cos|emb] @ W1 + b1)[i,f]
// ---------------------------------------------------------------------------
__global__ __launch_bounds__(256) void feat_kernel(
    const float* __restrict__ lpf, const float* __restrict__ Bm,
    const float* __restrict__ emb, const float* __restrict__ W1,
    const float* __restrict__ b1,
    float* __restrict__ P4, float* __restrict__ Y1T)
{
  __shared__ float sB[48];      // (3,16)
  __shared__ float sE[104];     // (13,8)
  __shared__ float sW[1280];    // (40,32)
  __shared__ float sb[32];
  const int tid = threadIdx.x;
  for (int t = tid; t < 48;   t += 256) sB[t] = Bm[t];
  for (int t = tid; t < 104;  t += 256) sE[t] = emb[t];
  for (int t = tid; t < 1280; t += 256) sW[t] = W1[t];
  if (tid < 32) sb[tid] = b1[tid];
  __syncthreads();

  const int i = blockIdx.x * 256 + tid;
  const float px = lpf[i*4+0], py = lpf[i*4+1], pz = lpf[i*4+2];
  const int   fr = (int)lpf[i*4+3];
  const float n2 = px*px + py*py + pz*pz;
  ((float4*)P4)[i] = make_float4(px, py, pz, n2);

  float xf[40];
  #pragma unroll
  for (int k = 0; k < 16; ++k) {
    const float t = 6.28318530717958647692f * (px*sB[k] + py*sB[16+k] + pz*sB[32+k]);
    xf[k]      = __sinf(t);
    xf[16 + k] = __cosf(t);
  }
  #pragma unroll
  for (int e = 0; e < 8; ++e) xf[32+e] = sE[fr*8 + e];

  #pragma unroll 4
  for (int f = 0; f < 32; ++f) {
    float acc = sb[f];
    #pragma unroll
    for (int k = 0; k < 40; ++k) acc += xf[k] * sW[k*32 + f];
    Y1T[f*N_NODES + i] = acc;       // f-major (coalesced across lanes)
  }
}

// ---------------------------------------------------------------------------
// Kernel 2: pass-1 pair phase. One block per 16-column i-tile; 8 waves split j.
// Per 16x16 j-tile: 1 dist-WMMA -> mask (C-layout) -> ballot bitmap store ->
// mask->B-layout (SWAPX16 + half-select) -> 8 aggregation WMMAs into
// G[f=0..31, i=0..15]. Tail: LDS reduce, h=relu(G/cnt), z = h@W2+b2 -> ZT.
// ---------------------------------------------------------------------------
__global__ __launch_bounds__(256) void agg1_kernel(
    const float* __restrict__ P4, const float* __restrict__ Y1T,
    const float* __restrict__ W2, const float* __restrict__ b2,
    float* __restrict__ ZT, float* __restrict__ CNT, unsigned* __restrict__ BM)
{
  __shared__ float red[16*33];   // [i][f0..31, 32=count]
  __shared__ float hbuf[16*32];
  const int tid  = threadIdx.x;
  const int lane = tid & 31;
  const int wave = tid >> 5;
  const int m    = lane & 15;
  const int half = lane >> 4;
  const int itile = blockIdx.x;
  const int i0    = itile * 16;

  for (int t = tid; t < 16*33; t += 256) red[t] = 0.f;
  __syncthreads();

  // B operand for the distance WMMA (fixed per block):
  // B[K,N=i]: vgpr0 = rows K0(-2x)/K2(-2z), vgpr1 = rows K1(-2y)/K3(1)
  const float4 q = ((const float4*)P4)[i0 + m];
  v2f bd;
  bd.x = half ? (-2.f*q.z) : (-2.f*q.x);
  bd.y = half ? 1.0f       : (-2.f*q.y);
  const float thr = 6.25f - q.w;   // dist^2 < 2.5^2  <=>  (|pj|^2 - 2pj.pi) < thr

  v8f g0 = {0.f,0.f,0.f,0.f,0.f,0.f,0.f,0.f};   // G[f=0..15,  i]
  v8f g1 = {0.f,0.f,0.f,0.f,0.f,0.f,0.f,0.f};   // G[f=16..31, i]
  float cnt = 0.f;

  for (int jt = wave; jt < NTILES; jt += 8) {
    const int jb = jt * 16;

    // A operand: A[M=j, K]: vgpr0 = K0(x)/K2(z), vgpr1 = K1(y)/K3(|p|^2)
    const float4 p = ((const float4*)P4)[jb + m];
    v2f ad;
    ad.x = half ? p.z : p.x;
    ad.y = half ? p.w : p.y;

    v8f zc = {0.f,0.f,0.f,0.f,0.f,0.f,0.f,0.f};
    v8f d2 = __builtin_amdgcn_wmma_f32_16x16x4_f32(
        false, ad, false, bd, (short)0, zc, false, false);

    // C layout: vgpr r = (j=jb+r, i=lane) lanes0-15 / (j=jb+r+8, i=lane-16) hi
    float mk[8]; unsigned bw[8];
    #pragma unroll
    for (int r = 0; r < 8; ++r) {
      const bool pr = d2[r] < thr;
      mk[r] = pr ? 1.f : 0.f;
      bw[r] = __builtin_amdgcn_ballot_w32(pr);
      cnt  += mk[r];
    }

    // bitmap: 8 uniform dwords per tile; lane r stores word r
    unsigned wv = bw[0];
    wv = (lane==1) ? bw[1] : wv;  wv = (lane==2) ? bw[2] : wv;
    wv = (lane==3) ? bw[3] : wv;  wv = (lane==4) ? bw[4] : wv;
    wv = (lane==5) ? bw[5] : wv;  wv = (lane==6) ? bw[6] : wv;
    wv = (lane==7) ? bw[7] : wv;
    if (lane < 8) BM[(size_t)itile * 4096u + (unsigned)(jt*8) + lane] = wv;

    // mask (C-layout, M=j) -> B operands (K=j rows): half-swap + select
    const float s0 = swap16(mk[0]), s1 = swap16(mk[1]);
    const float s2 = swap16(mk[2]), s3 = swap16(mk[3]);
    const float s4 = swap16(mk[4]), s5 = swap16(mk[5]);
    const float s6 = swap16(mk[6]), s7 = swap16(mk[7]);
    v2f Bc[4];
    Bc[0].x = half ? s2    : mk[0];  Bc[0].y = half ? s3    : mk[1];  // rows 0,2 / 1,3
    Bc[1].x = half ? s6    : mk[4];  Bc[1].y = half ? s7    : mk[5];  // rows 4,6 / 5,7
    Bc[2].x = half ? mk[2] : s0;     Bc[2].y = half ? mk[3] : s1;     // rows 8,10 / 9,11
    Bc[3].x = half ? mk[6] : s4;     Bc[3].y = half ? mk[7] : s5;     // rows 12,14 / 13,15

    // G[f,i] += Y1^T[f, j-chunk] * mask[j-chunk, i]; Y1T is f-major so the
    // two K-values per lane are contiguous -> single b64 load per operand.
    #pragma unroll
    for (int c = 0; c < 4; ++c) {
      const int j0 = jb + 4*c + 2*half;
      const float2 a0v = *(const float2*)(Y1T + (size_t)m        * N_NODES + j0);
      const float2 a1v = *(const float2*)(Y1T + (size_t)(16 + m) * N_NODES + j0);
      v2f A0; A0.x = a0v.x; A0.y = a0v.y;   // f-tile 0 (f = m)
      v2f A1; A1.x = a1v.x; A1.y = a1v.y;   // f-tile 1 (f = 16+m)
      g0 = __builtin_amdgcn_wmma_f32_16x16x4_f32(false, A0, false, Bc[c], (short)0, g0, false, false);
      g1 = __builtin_amdgcn_wmma_f32_16x16x4_f32(false, A1, false, Bc[c], (short)0, g1, false, false);
    }
  }

  // cross-wave reduction: G tiles are C-layout (f = r + 8*half, i = m)
  #pragma unroll
  for (int r = 0; r < 8; ++r) {
    const int f0 = r + 8*half;
    atomicAdd(&red[m*33 + f0],      g0[r]);
    atomicAdd(&red[m*33 + 16 + f0], g1[r]);
  }
  atomicAdd(&red[m*33 + 32], cnt);
  __syncthreads();

  // h = relu(G / (cnt + 1e-6)); persist cnt for pass 2
  for (int t = tid; t < 512; t += 256) {
    const int i = t >> 5, f = t & 31;
    const float c = red[i*33 + 32];
    hbuf[i*32 + f] = relu_f(red[i*33 + f] / (c + 1e-6f));
  }
  if (tid < 16) CNT[i0 + tid] = red[tid*33 + 32];
  __syncthreads();

  // z = h @ W2 + b2   (16 x 8, done in-block since the 16-row slab is complete)
  if (tid < 128) {
    const int i = tid >> 3, f = tid & 7;
    float acc = b2[f];
    #pragma unroll
    for (int k = 0; k < 32; ++k) acc += hbuf[i*32 + k] * W2[k*8 + f];
    ZT[f*N_NODES + (i0 + i)] = acc;   // f-major
  }
}

// ---------------------------------------------------------------------------
// Kernel 3: pass-2 pair phase. Rebuild masks from the bitmap, aggregate the
// 8-feature z with 4 WMMAs per 16x16 tile (f-rows 8..15 zero-padded).
// ---------------------------------------------------------------------------
__global__ __launch_bounds__(256) void agg2_kernel(
    const float* __restrict__ ZT, const float* __restrict__ CNT,
    const unsigned* __restrict__ BM, float* __restrict__ OUT)
{
  __shared__ float red[16*8];
  const int tid  = threadIdx.x;
  const int lane = tid & 31;
  const int wave = tid >> 5;
  const int m    = lane & 15;
  const int half = lane >> 4;
  const int itile = blockIdx.x;
  const int i0    = itile * 16;

  if (tid < 128) red[tid] = 0.f;
  __syncthreads();

  v8f g = {0.f,0.f,0.f,0.f,0.f,0.f,0.f,0.f};

  for (int jt = wave; jt < NTILES; jt += 8) {
    const int jb = jt * 16;
    const unsigned* wp = BM + (size_t)itile * 4096u + (unsigned)(jt*8);
    float mk[8];
    #pragma unroll
    for (int r = 0; r < 8; ++r) mk[r] = (float)((wp[r] >> lane) & 1u);

    const float s0 = swap16(mk[0]), s1 = swap16(mk[1]);
    const float s2 = swap16(mk[2]), s3 = swap16(mk[3]);
    const float s4 = swap16(mk[4]), s5 = swap16(mk[5]);
    const float s6 = swap16(mk[6]), s7 = swap16(mk[7]);
    v2f Bc[4];
    Bc[0].x = half ? s2    : mk[0];  Bc[0].y = half ? s3    : mk[1];
    Bc[1].x = half ? s6    : mk[4];  Bc[1].y = half ? s7    : mk[5];
    Bc[2].x = half ? mk[2] : s0;     Bc[2].y = half ? mk[3] : s1;
    Bc[3].x = half ? mk[6] : s4;     Bc[3].y = half ? mk[7] : s5;

    #pragma unroll
    for (int c = 0; c < 4; ++c) {
      const int j0 = jb + 4*c + 2*half;
      const float2 av = *(const float2*)(ZT + (size_t)(m & 7) * N_NODES + j0);
      float a0 = av.x, a1 = av.y;
      if (m >= 8) { a0 = 0.f; a1 = 0.f; }     // zero-pad f-rows 8..15
      v2f A; A.x = a0; A.y = a1;
      g = __builtin_amdgcn_wmma_f32_16x16x4_f32(false, A, false, Bc[c], (short)0, g, false, false);
    }
  }

  // only f-rows 0..7 (lanes 0..15 of each accumulator VGPR) are live
  #pragma unroll
  for (int r = 0; r < 8; ++r)
    if (half == 0) atomicAdd(&red[m*8 + r], g[r]);
  __syncthreads();

  if (tid < 128) {
    const int i = tid >> 3, f = tid & 7;
    const float c = CNT[i0 + i];
    OUT[(i0 + i)*8 + f] = relu_f(red[i*8 + f] / (c + 1e-6f));
  }
}

// ---------------------------------------------------------------------------
extern "C" void kernel_launch(void* const* d_in, const int* in_sizes, int n_in,
                              void* d_out, int out_size, void* d_ws, size_t ws_size,
                              hipStream_t stream)
{
  const float* lpf = (const float*)d_in[0];   // (8192,4)
  const float* Bm  = (const float*)d_in[1];   // (3,16)
  const float* emb = (const float*)d_in[2];   // (13,8)
  const float* W1  = (const float*)d_in[3];   // (40,32)
  const float* b1  = (const float*)d_in[4];   // (32,)
  const float* W2  = (const float*)d_in[5];   // (32,8)
  const float* b2  = (const float*)d_in[6];   // (8,)

  char* ws = (char*)d_ws;
  float*    P4  = (float*)(ws + WS_P4);
  float*    Y1T = (float*)(ws + WS_Y1T);
  float*    ZT  = (float*)(ws + WS_ZT);
  float*    CNT = (float*)(ws + WS_CNT);
  unsigned* BM  = (unsigned*)(ws + WS_BM);
  float*    OUT = (float*)d_out;

  feat_kernel<<<N_NODES/256, 256, 0, stream>>>(lpf, Bm, emb, W1, b1, P4, Y1T);
  agg1_kernel<<<NTILES, 256, 0, stream>>>(P4, Y1T, W2, b2, ZT, CNT, BM);
  agg2_kernel<<<NTILES, 256, 0, stream>>>(ZT, CNT, BM, OUT);
}